// LaplacianPyramid_57758720197011
// MI455X (gfx1250) — compile-verified
//
#include <hip/hip_runtime.h>

typedef float v4f __attribute__((ext_vector_type(4)));
typedef float v2f __attribute__((ext_vector_type(2)));

// Bilinear tap of one square NxN layer at uv-space (ux,uy), zeros padding,
// align_corners=False. Uses 4 dword gathers (2 adjacent per row) with a
// clamped address and branch-free border selects.
__device__ __forceinline__ float sample_one(const float* __restrict__ img,
                                            const int N, float ux, float uy) {
  const float fN = (float)N;
  float x = ux * fN - 0.5f;              // == (grid_x+1)*(N/2) - 0.5
  float y = uy * fN - 0.5f;
  float xf = floorf(x), yf = floorf(y);
  float wx1 = x - xf, wy1 = y - yf;
  float wx0 = 1.0f - wx1, wy0 = 1.0f - wy1;
  int x0 = (int)xf, y0 = (int)yf;

  // x: load the adjacent pair at clamped xc; re-route values at the border.
  int xc = min(max(x0, 0), N - 2);
  bool straight = (x0 == xc);            // false only at x0==-1 or x0==N-1
  bool lv = (x0 >= 0) && (x0 < N);       // left tap (x0) in range
  bool rv = (x0 >= -1) && (x0 < N - 1);  // right tap (x0+1) in range

  // y: clamp row index, zero the row weight if out of range.
  int y0c = min(max(y0, 0), N - 1);
  int y1c = min(max(y0 + 1, 0), N - 1);
  float vy0 = ((y0 >= 0) && (y0 < N)) ? wy0 : 0.0f;
  float vy1 = ((y0 >= -1) && (y0 < N - 1)) ? wy1 : 0.0f;

  const float* __restrict__ r0 = img + y0c * N + xc;
  const float* __restrict__ r1 = img + y1c * N + xc;
  float v00 = r0[0];
  float v01 = r0[1];
  float v10 = r1[0];
  float v11 = r1[1];

  float l0 = lv ? (straight ? v00 : v01) : 0.0f;  // x0==N-1 -> pair[1]
  float g0 = rv ? (straight ? v01 : v00) : 0.0f;  // x0==-1  -> pair[0]
  float l1 = lv ? (straight ? v10 : v11) : 0.0f;
  float g1 = rv ? (straight ? v11 : v10) : 0.0f;

  float row0 = wx0 * l0 + wx1 * g0;
  float row1 = wx0 * l1 + wx1 * g1;
  return vy0 * row0 + vy1 * row1;
}

__device__ __forceinline__ float sample_pyramid(const float* __restrict__ L1,
                                                const float* __restrict__ L2,
                                                const float* __restrict__ L3,
                                                const float* __restrict__ L4,
                                                float ux, float uy) {
  float s = sample_one(L1, 2048, ux, uy);
  s += sample_one(L2, 1024, ux, uy);
  s += sample_one(L3, 512, ux, uy);
  s += sample_one(L4, 256, ux, uy);
  return s;
}

__global__ void __launch_bounds__(256)
grid_sample_pyr_kernel(const float* __restrict__ uv,
                       const float* __restrict__ L1,
                       const float* __restrict__ L2,
                       const float* __restrict__ L3,
                       const float* __restrict__ L4,
                       float* __restrict__ out,
                       int nPairs) {  // one "pair" = 2 output pixels
  const int T = gridDim.x * blockDim.x;
  const v4f* __restrict__ uv4 = (const v4f*)uv;
  v2f* __restrict__ out2 = (v2f*)out;

  for (int i = blockIdx.x * blockDim.x + threadIdx.x; i < nPairs; i += T) {
    // Prefetch next grid-stride iteration's uv (stream) while gathers run.
    if (i + T < nPairs)
      __builtin_prefetch((const void*)(uv4 + (i + T)), 0, 1);

    // Stream-once data: non-temporal so it doesn't evict the L2-resident
    // 22 MB texture working set (MI455X L2 = 192 MB).
    v4f u = __builtin_nontemporal_load(uv4 + i);

    float s0 = sample_pyramid(L1, L2, L3, L4, u.x, u.y);  // pixel 2i
    float s1 = sample_pyramid(L1, L2, L3, L4, u.z, u.w);  // pixel 2i+1

    v2f r;
    r.x = s0;
    r.y = s1;
    __builtin_nontemporal_store(r, out2 + i);
  }
}

extern "C" void kernel_launch(void* const* d_in, const int* in_sizes, int n_in,
                              void* d_out, int out_size, void* d_ws, size_t ws_size,
                              hipStream_t stream) {
  const float* uv = (const float*)d_in[0];
  const float* L1 = (const float*)d_in[1];
  const float* L2 = (const float*)d_in[2];
  const float* L3 = (const float*)d_in[3];
  const float* L4 = (const float*)d_in[4];
  float* out = (float*)d_out;

  const int nPairs = out_size / 2;            // 4,194,304 for 8x1024x1024
  const int threads = 256;                    // 8 wave32 per block
  const int itersPerThread = 4;               // grid-stride depth for prefetch
  int blocks = (nPairs + threads * itersPerThread - 1) / (threads * itersPerThread);
  if (blocks < 1) blocks = 1;

  grid_sample_pyr_kernel<<<dim3(blocks), dim3(threads), 0, stream>>>(
      uv, L1, L2, L3, L4, out, nPairs);
}